// MultLayerAdaptiveSimple_42013370089772
// MI455X (gfx1250) — compile-verified
//
#include <hip/hip_runtime.h>
#include <stdint.h>

// out[b,s,d] = X[b,s,d] * W[r(b,s)][0] + Y[b,s,d] * W[r(b,s)][1]
// B=4, S=4096, D=2048 fp32. HBM-bound: ~384MB moved, floor ~16.5us @ 23.3TB/s.
// Uses gfx1250 async global->LDS copies (ASYNCcnt) as a double-buffered
// software pipeline; NT temporal hints keep the 384MB stream out of L2.

typedef float v4f __attribute__((ext_vector_type(4)));

// D = 2048 floats per row = 512 float4 chunks per row -> row = chunk >> 9
#define ROW_SHIFT_V4 9
#define CPT 4            // chunks (float4) per thread
#define BT  256          // threads per block (8 wave32)

// ---------------------------------------------------------------------------
// Async double-buffered pipeline kernel. Requires n_chunks % (BT*CPT) == 0
// (wave-uniform ASYNCcnt bookkeeping -> no divergent tail allowed).
// ---------------------------------------------------------------------------
__global__ __launch_bounds__(BT)
void blend_rows_async_kernel(const v4f* __restrict__ X,
                             const v4f* __restrict__ Y,
                             const int* __restrict__ reward_lo, // int64 viewed as int pairs
                             const float* __restrict__ W,
                             v4f* __restrict__ out,
                             int n_chunks)
{
    __shared__ v4f sx[2][BT];
    __shared__ v4f sy[2][BT];

    const float w00 = W[0], w01 = W[1], w10 = W[2], w11 = W[3];

    const int t      = threadIdx.x;
    const int tid    = blockIdx.x * BT + t;
    const int stride = gridDim.x * BT;

    // Per-lane LDS byte addresses for both buffers.
    const uint32_t lx0 = (uint32_t)(uintptr_t)&sx[0][t];
    const uint32_t lx1 = (uint32_t)(uintptr_t)&sx[1][t];
    const uint32_t ly0 = (uint32_t)(uintptr_t)&sy[0][t];
    const uint32_t ly1 = (uint32_t)(uintptr_t)&sy[1][t];

    // Prologue: stage iteration 0 tiles into buffer 0.
    {
        const uint64_t gx = (uint64_t)(uintptr_t)&X[tid];
        const uint64_t gy = (uint64_t)(uintptr_t)&Y[tid];
        asm volatile("global_load_async_to_lds_b128 %0, %1, off th:TH_LOAD_NT"
                     :: "v"(lx0), "v"(gx) : "memory");
        asm volatile("global_load_async_to_lds_b128 %0, %1, off th:TH_LOAD_NT"
                     :: "v"(ly0), "v"(gy) : "memory");
    }

#pragma unroll
    for (int k = 0; k < CPT; ++k) {
        const int c = tid + k * stride;

        if (k + 1 < CPT) {
            // Prefetch next iteration's tiles into the other buffer.
            const int cn = tid + (k + 1) * stride;
            const uint64_t gx = (uint64_t)(uintptr_t)&X[cn];
            const uint64_t gy = (uint64_t)(uintptr_t)&Y[cn];
            const uint32_t lxn = ((k + 1) & 1) ? lx1 : lx0;
            const uint32_t lyn = ((k + 1) & 1) ? ly1 : ly0;
            asm volatile("global_load_async_to_lds_b128 %0, %1, off th:TH_LOAD_NT"
                         :: "v"(lxn), "v"(gx) : "memory");
            asm volatile("global_load_async_to_lds_b128 %0, %1, off th:TH_LOAD_NT"
                         :: "v"(lyn), "v"(gy) : "memory");
            // Async loads complete in order: leaving 2 outstanding means the
            // current buffer's copies have landed in LDS.
            asm volatile("s_wait_asynccnt 0x2" ::: "memory");
        } else {
            asm volatile("s_wait_asynccnt 0x0" ::: "memory");
        }

        // Per-row weight select (reward is 128KB total -> cache-resident).
        const int row = c >> ROW_SHIFT_V4;
        const int idx = reward_lo[2 * row];
        const float a = idx ? w10 : w00;
        const float b = idx ? w11 : w01;

        // Consume current buffer (ds_load_b128), compute, stream out.
        const v4f x = (k & 1) ? sx[1][t] : sx[0][t];
        const v4f y = (k & 1) ? sy[1][t] : sy[0][t];
        v4f r = x * a + y * b;
        __builtin_nontemporal_store(r, &out[c]);
    }
}

// ---------------------------------------------------------------------------
// Fallback direct-load kernel (any shape): 128-bit NT loads/stores + pk_fma.
// ---------------------------------------------------------------------------
__global__ __launch_bounds__(BT)
void blend_rows_direct_kernel(const v4f* __restrict__ X,
                              const v4f* __restrict__ Y,
                              const int* __restrict__ reward_lo,
                              const float* __restrict__ W,
                              v4f* __restrict__ out,
                              int n_chunks)
{
    const float w00 = W[0], w01 = W[1], w10 = W[2], w11 = W[3];
    const int tid    = blockIdx.x * BT + threadIdx.x;
    const int stride = gridDim.x * BT;

#pragma unroll
    for (int k = 0; k < CPT; ++k) {
        const int c = tid + k * stride;
        if (c < n_chunks) {
            const int row = c >> ROW_SHIFT_V4;
            const int idx = reward_lo[2 * row];
            const float a = idx ? w10 : w00;
            const float b = idx ? w11 : w01;
            const v4f x = __builtin_nontemporal_load(&X[c]);
            const v4f y = __builtin_nontemporal_load(&Y[c]);
            v4f r = x * a + y * b;
            __builtin_nontemporal_store(r, &out[c]);
        }
    }
}

extern "C" void kernel_launch(void* const* d_in, const int* in_sizes, int n_in,
                              void* d_out, int out_size, void* d_ws, size_t ws_size,
                              hipStream_t stream)
{
    const v4f*   X   = (const v4f*)d_in[0];
    const v4f*   Y   = (const v4f*)d_in[1];
    const int*   rwd = (const int*)d_in[2];   // int64 reward, read low dwords
    const float* W   = (const float*)d_in[3];
    v4f*         out = (v4f*)d_out;

    const int n_chunks = out_size / 4;        // float4 chunks
    const int per_block = BT * CPT;

    if (n_chunks % per_block == 0) {
        // Exact tiling (true for B=4,S=4096,D=2048 -> 8192 blocks):
        // async double-buffered pipeline.
        const int blocks = n_chunks / per_block;
        blend_rows_async_kernel<<<blocks, BT, 0, stream>>>(X, Y, rwd, W, out, n_chunks);
    } else {
        const int threads_needed = (n_chunks + CPT - 1) / CPT;
        const int blocks = (threads_needed + BT - 1) / BT;
        blend_rows_direct_kernel<<<blocks, BT, 0, stream>>>(X, Y, rwd, W, out, n_chunks);
    }
}